// bond_agg_layer_77197742178842
// MI455X (gfx1250) — compile-verified
//
#include <hip/hip_runtime.h>
#include <hip/hip_bf16.h>

typedef float v2f __attribute__((ext_vector_type(2)));
typedef float v8f __attribute__((ext_vector_type(8)));

#define DDIM 128
#define NSEG 131072

// Order-preserving float -> uint key so atomicMax(u32) implements float max
// (handles negatives). key=0 corresponds to "minus infinity-ish" sentinel.
__device__ __forceinline__ unsigned f2key(float f) {
    unsigned b = __float_as_uint(f);
    return (b & 0x80000000u) ? ~b : (b | 0x80000000u);
}
__device__ __forceinline__ float key2f(unsigned k) {
    unsigned b = (k & 0x80000000u) ? (k & 0x7FFFFFFFu) : ~k;
    return __uint_as_float(b);
}

// Kernel 1: s[e] = leaky_relu(x[e] . w_atten), per-segment max via atomic key.
// One wave (32 lanes) per edge; lane l covers columns 4l..4l+3.
__global__ void attn_score_kernel(const float* __restrict__ x,
                                  const float* __restrict__ w_atten,
                                  const int* __restrict__ seg,
                                  float* __restrict__ s,
                                  unsigned* __restrict__ smaxkey, int E) {
    int lane  = threadIdx.x & 31;
    int wave  = blockIdx.x * (blockDim.x >> 5) + (threadIdx.x >> 5);
    int nwave = gridDim.x * (blockDim.x >> 5);
    float4 wv = reinterpret_cast<const float4*>(w_atten)[lane];
    for (int e = wave; e < E; e += nwave) {
        float4 xv = reinterpret_cast<const float4*>(x + (size_t)e * DDIM)[lane];
        float d = xv.x * wv.x + xv.y * wv.y + xv.z * wv.z + xv.w * wv.w;
#pragma unroll
        for (int off = 16; off > 0; off >>= 1) d += __shfl_xor(d, off, 32);
        if (lane == 0) {
            float sv = d > 0.0f ? d : 0.01f * d;
            s[e] = sv;
            atomicMax(&smaxkey[seg[e]], f2key(sv));
        }
    }
}

// Kernel 2: denom[seg] += exp(s - smax), cnt[seg] += 1
__global__ void denom_kernel(const float* __restrict__ s,
                             const int* __restrict__ seg,
                             const unsigned* __restrict__ smaxkey,
                             float* __restrict__ denom,
                             float* __restrict__ cnt, int E) {
    int e = blockIdx.x * blockDim.x + threadIdx.x;
    if (e >= E) return;
    int sg = seg[e];
    float ev = expf(s[e] - key2f(smaxkey[sg]));
    atomicAdd(&denom[sg], ev);
    atomicAdd(&cnt[sg], 1.0f);
}

// Kernel 3: ssum[seg] += atten * x[e]; wave per edge, float4 per lane.
__global__ void scatter_kernel(const float* __restrict__ x,
                               const float* __restrict__ s,
                               const int* __restrict__ seg,
                               const unsigned* __restrict__ smaxkey,
                               const float* __restrict__ denom,
                               float* __restrict__ ssum, int E) {
    int lane  = threadIdx.x & 31;
    int wave  = blockIdx.x * (blockDim.x >> 5) + (threadIdx.x >> 5);
    int nwave = gridDim.x * (blockDim.x >> 5);
    for (int e = wave; e < E; e += nwave) {
        int sg = seg[e];
        float a = expf(s[e] - key2f(smaxkey[sg])) / denom[sg];
        float4 xv = reinterpret_cast<const float4*>(x + (size_t)e * DDIM)[lane];
        float* dst = ssum + (size_t)sg * DDIM + lane * 4;
        atomicAdd(dst + 0, a * xv.x);
        atomicAdd(dst + 1, a * xv.y);
        atomicAdd(dst + 2, a * xv.z);
        atomicAdd(dst + 3, a * xv.w);
    }
}

// Kernel 4: bond = ssum / max(cnt,1); h = (bond @ w_lin.T) @ w_lin2.T via
// fp32 WMMA 16x16x4; accumulate per-column sum / sumsq for batchnorm.
// 128 threads = 4 waves; each wave owns a 16-row band (no cross-wave deps).
__global__ void __launch_bounds__(128)
gemm_kernel(const float* __restrict__ ssum, const float* __restrict__ cnt,
            const float* __restrict__ w_lin, const float* __restrict__ w_lin2,
            float* __restrict__ h, float* __restrict__ colsum,
            float* __restrict__ colsumsq) {
    __shared__ float lbond[4][16][132];  // padded: avoid 64-bank conflicts
    __shared__ float lh1[4][16][132];
    int lane = threadIdx.x & 31;
    int wave = threadIdx.x >> 5;
    int m0 = (blockIdx.x * 4 + wave) * 16;

    // Load 16x128 bond tile into LDS (scatter-mean finalize).
    for (int r = 0; r < 16; ++r) {
        int row = m0 + r;
        float inv = 1.0f / fmaxf(cnt[row], 1.0f);
        float4 v = reinterpret_cast<const float4*>(ssum + (size_t)row * DDIM)[lane];
        lbond[wave][r][lane * 4 + 0] = v.x * inv;
        lbond[wave][r][lane * 4 + 1] = v.y * inv;
        lbond[wave][r][lane * 4 + 2] = v.z * inv;
        lbond[wave][r][lane * 4 + 3] = v.w * inv;
    }

    int koff  = (lane >> 4) << 1;  // A/B: hi-half lanes carry K+2,K+3
    int rrow  = lane & 15;         // A: row M, B: col N
    int rbase = (lane >> 4) << 3;  // D: hi-half lanes carry M+8..M+15

    // GEMM1: h1 = bond @ w_lin.T   (B[k][n] = w_lin[n*D + k])
    for (int jt = 0; jt < 8; ++jt) {
        v8f acc = {};
        for (int k0 = 0; k0 < DDIM; k0 += 4) {
            v2f a, b;
            a[0] = lbond[wave][rrow][k0 + koff];
            a[1] = lbond[wave][rrow][k0 + koff + 1];
            const float* wp = w_lin + (size_t)(jt * 16 + rrow) * DDIM + k0 + koff;
            b[0] = wp[0]; b[1] = wp[1];
            acc = __builtin_amdgcn_wmma_f32_16x16x4_f32(false, a, false, b,
                                                        (short)0, acc, false, false);
        }
#pragma unroll
        for (int r = 0; r < 8; ++r)
            lh1[wave][rbase + r][jt * 16 + rrow] = acc[r];
    }

    // GEMM2: h = h1 @ w_lin2.T; emit h + per-column batchnorm partials.
    for (int jt = 0; jt < 8; ++jt) {
        v8f acc = {};
        for (int k0 = 0; k0 < DDIM; k0 += 4) {
            v2f a, b;
            a[0] = lh1[wave][rrow][k0 + koff];
            a[1] = lh1[wave][rrow][k0 + koff + 1];
            const float* wp = w_lin2 + (size_t)(jt * 16 + rrow) * DDIM + k0 + koff;
            b[0] = wp[0]; b[1] = wp[1];
            acc = __builtin_amdgcn_wmma_f32_16x16x4_f32(false, a, false, b,
                                                        (short)0, acc, false, false);
        }
        float ps = 0.0f, pq = 0.0f;
#pragma unroll
        for (int r = 0; r < 8; ++r) {
            float v = acc[r];
            h[(size_t)(m0 + rbase + r) * DDIM + jt * 16 + rrow] = v;
            ps += v; pq += v * v;
        }
        ps += __shfl_xor(ps, 16, 32);   // combine lo/hi half-rows of the tile
        pq += __shfl_xor(pq, 16, 32);
        if (lane < 16) {
            atomicAdd(&colsum[jt * 16 + lane], ps);
            atomicAdd(&colsumsq[jt * 16 + lane], pq);
        }
    }
}

// Kernel 5: finalize batch statistics (biased variance, eps 1e-5).
__global__ void stats_kernel(const float* __restrict__ colsum,
                             const float* __restrict__ colsumsq,
                             float* __restrict__ mu, float* __restrict__ invstd) {
    int j = threadIdx.x;
    float m = colsum[j] * (1.0f / (float)NSEG);
    float v = colsumsq[j] * (1.0f / (float)NSEG) - m * m;
    mu[j] = m;
    invstd[j] = rsqrtf(v + 1e-5f);
}

// Kernel 6: out = gelu_exact((h - mu) * invstd * gamma + beta)
__global__ void norm_gelu_kernel(const float* __restrict__ h,
                                 const float* __restrict__ mu,
                                 const float* __restrict__ invstd,
                                 const float* __restrict__ gamma,
                                 const float* __restrict__ beta,
                                 float* __restrict__ out) {
    size_t i = (size_t)blockIdx.x * blockDim.x + threadIdx.x;
    int j = (int)(i & (DDIM - 1));
    float t = (h[i] - mu[j]) * invstd[j] * gamma[j] + beta[j];
    out[i] = 0.5f * t * (1.0f + erff(t * 0.7071067811865475f));
}

extern "C" void kernel_launch(void* const* d_in, const int* in_sizes, int n_in,
                              void* d_out, int out_size, void* d_ws, size_t ws_size,
                              hipStream_t stream) {
    const float* x       = (const float*)d_in[0];
    const float* w_atten = (const float*)d_in[1];
    const float* w_lin   = (const float*)d_in[2];
    const float* w_lin2  = (const float*)d_in[3];
    const float* gamma   = (const float*)d_in[4];
    const float* beta    = (const float*)d_in[5];
    const int*   eidx    = (const int*)d_in[6];   // [2, E], row 1 = segments
    int E = in_sizes[0] / DDIM;
    const int* seg = eidx + E;

    // Carve workspace (256B-aligned slabs).
    char* ws = (char*)d_ws;
    size_t off = 0;
    auto carve = [&](size_t bytes) -> void* {
        void* p = ws + off;
        off += (bytes + 255) & ~(size_t)255;
        return p;
    };
    float*    s        = (float*)carve((size_t)E * 4);
    unsigned* smaxkey  = (unsigned*)carve((size_t)NSEG * 4);  // zeroed below
    float*    denom    = (float*)carve((size_t)NSEG * 4);     // zeroed
    float*    cnt      = (float*)carve((size_t)NSEG * 4);     // zeroed
    float*    colsum   = (float*)carve(DDIM * 4);             // zeroed
    float*    colsumsq = (float*)carve(DDIM * 4);             // zeroed
    float*    mu       = (float*)carve(DDIM * 4);
    float*    invstd   = (float*)carve(DDIM * 4);
    float*    ssum     = (float*)carve((size_t)NSEG * DDIM * 4); // zeroed
    float*    h        = (float*)carve((size_t)NSEG * DDIM * 4);

    // Zero everything from smaxkey through end of ssum in one async memset
    // (mu/invstd sit inside the range; they are fully overwritten later).
    size_t zbytes = (size_t)((char*)ssum + (size_t)NSEG * DDIM * 4 - (char*)smaxkey);
    hipMemsetAsync(smaxkey, 0, zbytes, stream);

    attn_score_kernel<<<4096, 256, 0, stream>>>(x, w_atten, seg, s, smaxkey, E);
    denom_kernel<<<(E + 255) / 256, 256, 0, stream>>>(s, seg, smaxkey, denom, cnt, E);
    scatter_kernel<<<4096, 256, 0, stream>>>(x, s, seg, smaxkey, denom, ssum, E);
    gemm_kernel<<<NSEG / 64, 128, 0, stream>>>(ssum, cnt, w_lin, w_lin2,
                                               h, colsum, colsumsq);
    stats_kernel<<<1, DDIM, 0, stream>>>(colsum, colsumsq, mu, invstd);
    norm_gelu_kernel<<<(size_t)NSEG * DDIM / 256, 256, 0, stream>>>(
        h, mu, invstd, gamma, beta, (float*)d_out);
}